// netVLADlayer_1030792151613
// MI455X (gfx1250) — compile-verified
//
#include <hip/hip_runtime.h>

typedef __attribute__((ext_vector_type(2))) float v2f;
typedef __attribute__((ext_vector_type(8))) float v8f;

#define EPS      1e-12f
#define N_       32
#define HW_      1024      // 32*32
#define D_       512
#define K_       64
#define OUT_     4096
#define NPIX     (N_*HW_)  // 32768
#define KD_      (K_*D_)   // 32768
#define CHUNKS   16
#define CHUNK_K  (KD_/CHUNKS)   // 2048

__device__ __forceinline__ v8f wmma_f32_k4(v2f a, v2f b, v8f c) {
  // D = A(16x4) * B(4x16) + C, full f32. 8-arg VOP3P form:
  // (neg_a, A, neg_b, B, c_mod, C, reuse_a, reuse_b)
  return __builtin_amdgcn_wmma_f32_16x16x4_f32(false, a, false, b, (short)0, c,
                                               false, false);
}

// ---------------- 1) logits = x @ W1 + b1  -> s ----------------
// x:[NPIX,D]  W1:[D,K]  s:[NPIX,K].  One 16x16 tile per wave.
__global__ void k_logits(const float* __restrict__ x, const float* __restrict__ w1,
                         const float* __restrict__ b1, float* __restrict__ s) {
  int wid  = (blockIdx.x * blockDim.x + threadIdx.x) >> 5;
  int lane = threadIdx.x & 31;
  int mtile = wid >> 2;            // 0..2047 (pixel tiles)
  int ntile = wid & 3;             // 0..3    (cluster tiles)
  int half = lane >> 4, l16 = lane & 15, koff = half * 2;
  int arow = mtile * 16 + l16;     // pixel
  int bcol = ntile * 16 + l16;     // cluster
  v8f acc = {};
  for (int k0 = 0; k0 < D_; k0 += 4) {
    v2f a = *(const v2f*)(x + (size_t)arow * D_ + k0 + koff);   // 8B-aligned
    v2f b;
    b.x = w1[(size_t)(k0 + koff) * K_ + bcol];
    b.y = w1[(size_t)(k0 + koff + 1) * K_ + bcol];
    acc = wmma_f32_k4(a, b, acc);
  }
  float bias = b1[bcol];
#pragma unroll
  for (int r = 0; r < 8; ++r) {
    int m = mtile * 16 + r + half * 8;
    s[(size_t)m * K_ + bcol] = acc[r] + bias;
  }
}

// ---------------- 2) softmax over K=64, wave per row ----------------
__global__ void k_softmax(float* __restrict__ s) {
  int wid  = (blockIdx.x * blockDim.x + threadIdx.x) >> 5;
  int lane = threadIdx.x & 31;
  float* row = s + (size_t)wid * K_;
  float a = row[lane], b = row[lane + 32];
  float m = fmaxf(a, b);
  for (int off = 16; off > 0; off >>= 1) m = fmaxf(m, __shfl_xor(m, off, 32));
  float ea = __expf(a - m), eb = __expf(b - m);
  float sum = ea + eb;
  for (int off = 16; off > 0; off >>= 1) sum += __shfl_xor(sum, off, 32);
  float inv = 1.0f / sum;
  row[lane] = ea * inv;
  row[lane + 32] = eb * inv;
}

// ---------------- 3) sum_s[n,k] = sum_hw s ----------------
__global__ void k_sums(const float* __restrict__ s, float* __restrict__ sum_s) {
  int t = blockIdx.x * blockDim.x + threadIdx.x;   // 0..2047
  int n = t >> 6, k = t & 63;
  const float* base = s + (size_t)n * HW_ * K_ + k;
  float acc = 0.f;
  for (int hw = 0; hw < HW_; ++hw) acc += base[(size_t)hw * K_];
  sum_s[t] = acc;
}

// ---------------- 4) v[n,k,d] = s_n^T @ x_n + sum_s * C^T ----------------
__global__ void k_vlad(const float* __restrict__ s, const float* __restrict__ x,
                       const float* __restrict__ sum_s, const float* __restrict__ Cb,
                       float* __restrict__ v) {
  int wid  = (blockIdx.x * blockDim.x + threadIdx.x) >> 5;   // 0..4095
  int lane = threadIdx.x & 31;
  int n = wid >> 7;
  int rem = wid & 127;
  int mtile = rem >> 5;    // cluster tile 0..3
  int dtile = rem & 31;    // d tile 0..31
  int half = lane >> 4, l16 = lane & 15, koff = half * 2;
  int kcl_a = mtile * 16 + l16;   // A row = cluster
  int dcol  = dtile * 16 + l16;   // B col = d
  const float* sn = s + (size_t)n * HW_ * K_;
  const float* xn = x + (size_t)n * HW_ * D_;
  v8f acc = {};
  for (int hw0 = 0; hw0 < HW_; hw0 += 4) {
    int h = hw0 + koff;
    v2f a, b;
    a.x = sn[(size_t)h * K_ + kcl_a];
    a.y = sn[(size_t)(h + 1) * K_ + kcl_a];
    b.x = xn[(size_t)h * D_ + dcol];
    b.y = xn[(size_t)(h + 1) * D_ + dcol];
    acc = wmma_f32_k4(a, b, acc);
  }
#pragma unroll
  for (int r = 0; r < 8; ++r) {
    int kcl = mtile * 16 + r + half * 8;
    float val = acc[r] + sum_s[n * K_ + kcl] * Cb[(size_t)dcol * K_ + kcl];
    v[(size_t)(n * K_ + kcl) * D_ + dcol] = val;
  }
}

// ---------------- 5) intra-cluster L2 over D, accumulate per-n sumsq ----------------
__global__ void k_intranorm(float* __restrict__ v, float* __restrict__ gnorm) {
  int wid  = (blockIdx.x * blockDim.x + threadIdx.x) >> 5;   // 0..2047 = (n,k)
  int lane = threadIdx.x & 31;
  int n = wid >> 6;
  float* row = v + (size_t)wid * D_;
  float vals[16];
  float ss = 0.f;
#pragma unroll
  for (int i = 0; i < 16; ++i) { float t = row[lane + i * 32]; vals[i] = t; ss += t * t; }
  for (int off = 16; off > 0; off >>= 1) ss += __shfl_xor(ss, off, 32);
  float inv = rsqrtf(fmaxf(ss, EPS));
#pragma unroll
  for (int i = 0; i < 16; ++i) row[lane + i * 32] = vals[i] * inv;
  if (lane == 0) atomicAdd(&gnorm[n], ss * inv * inv);
}

// ---------------- 6) per-n global L2 scale (in place) ----------------
__global__ void k_gscale(float* __restrict__ v, const float* __restrict__ gnorm) {
  int i = blockIdx.x * blockDim.x + threadIdx.x;   // 0..N*KD-1
  int n = i >> 15;                                 // /32768
  v[i] *= rsqrtf(fmaxf(gnorm[n], EPS));
}

// ---------------- 7) projection, split-K into 16 chunks ----------------
// vbar:[32,KD]  w2:[KD,OUT]  part:[CHUNKS,32,OUT]. One wave does both M-tiles.
__global__ void k_proj(const float* __restrict__ vbar, const float* __restrict__ w2,
                       float* __restrict__ part) {
  int wid  = (blockIdx.x * blockDim.x + threadIdx.x) >> 5;   // 0..4095
  int lane = threadIdx.x & 31;
  int chunk = wid >> 8;    // 0..15
  int ntile = wid & 255;   // 0..255
  int half = lane >> 4, l16 = lane & 15, koff = half * 2;
  int ncol = ntile * 16 + l16;
  const float* a0p = vbar + (size_t)l16 * KD_;         // rows 0..15
  const float* a1p = vbar + (size_t)(16 + l16) * KD_;  // rows 16..31
  v8f acc0 = {}, acc1 = {};
  int kbeg = chunk * CHUNK_K, kend = kbeg + CHUNK_K;
  for (int k0 = kbeg; k0 < kend; k0 += 4) {
    int k = k0 + koff;
    v2f a0 = *(const v2f*)(a0p + k);
    v2f a1 = *(const v2f*)(a1p + k);
    v2f b;
    b.x = w2[(size_t)k * OUT_ + ncol];
    b.y = w2[(size_t)(k + 1) * OUT_ + ncol];
    acc0 = wmma_f32_k4(a0, b, acc0);   // B reused by both WMMAs
    acc1 = wmma_f32_k4(a1, b, acc1);
  }
  float* pc = part + (size_t)chunk * (N_ * OUT_);
#pragma unroll
  for (int r = 0; r < 8; ++r) {
    int m0 = r + half * 8;
    pc[(size_t)m0 * OUT_ + ncol]        = acc0[r];
    pc[(size_t)(m0 + 16) * OUT_ + ncol] = acc1[r];
  }
}

// ---------------- 8) split-K reduce + bias + global sumsq ----------------
__global__ void k_finish(const float* __restrict__ part, const float* __restrict__ b2,
                         float* __restrict__ out, float* __restrict__ sumsq) {
  int i = blockIdx.x * blockDim.x + threadIdx.x;   // 0..131071
  int ncol = i & (OUT_ - 1);
  float acc = b2[ncol];
#pragma unroll
  for (int c = 0; c < CHUNKS; ++c) acc += part[(size_t)c * (N_ * OUT_) + i];
  out[i] = acc;
  float ss = acc * acc;
  for (int off = 16; off > 0; off >>= 1) ss += __shfl_xor(ss, off, 32);
  if ((threadIdx.x & 31) == 0) atomicAdd(sumsq, ss);
}

// ---------------- 9) global L2 over whole [32,4096] ----------------
__global__ void k_outscale(float* __restrict__ out, const float* __restrict__ sumsq) {
  int i = blockIdx.x * blockDim.x + threadIdx.x;
  out[i] *= rsqrtf(fmaxf(sumsq[0], EPS));
}

extern "C" void kernel_launch(void* const* d_in, const int* in_sizes, int n_in,
                              void* d_out, int out_size, void* d_ws, size_t ws_size,
                              hipStream_t stream) {
  (void)in_sizes; (void)n_in; (void)out_size; (void)ws_size;
  const float* x  = (const float*)d_in[0];   // [32,32,32,512]
  const float* w1 = (const float*)d_in[1];   // [512,64]
  const float* b1 = (const float*)d_in[2];   // [64]
  const float* Cb = (const float*)d_in[3];   // [512,64]
  const float* w2 = (const float*)d_in[4];   // [32768,4096]
  const float* b2 = (const float*)d_in[5];   // [4096]
  float* out = (float*)d_out;                // [32,4096]

  float* ws = (float*)d_ws;
  float* s     = ws;                         // 32768*64   = 2,097,152
  float* sum_s = s     + (size_t)NPIX * K_;  // 2048
  float* v     = sum_s + (size_t)N_ * K_;    // 32*64*512  = 1,048,576
  float* gnorm = v     + (size_t)N_ * KD_;   // 32
  float* part  = gnorm + N_;                 // 16*32*4096 = 2,097,152
  float* sumsq = part  + (size_t)CHUNKS * N_ * OUT_;  // 1

  hipMemsetAsync(gnorm, 0, (N_ + 1) * sizeof(float) + 0, stream); // gnorm[32]
  hipMemsetAsync(sumsq, 0, sizeof(float), stream);

  k_logits   <<<1024, 256, 0, stream>>>(x, w1, b1, s);       // 8192 waves
  k_softmax  <<<4096, 256, 0, stream>>>(s);                  // 32768 waves
  k_sums     <<<8,    256, 0, stream>>>(s, sum_s);           // 2048 threads
  k_vlad     <<<512,  256, 0, stream>>>(s, x, sum_s, Cb, v); // 4096 waves
  k_intranorm<<<256,  256, 0, stream>>>(v, gnorm);           // 2048 waves
  k_gscale   <<<4096, 256, 0, stream>>>(v, gnorm);           // 1M elems
  k_proj     <<<512,  256, 0, stream>>>(v, w2, part);        // 4096 waves
  k_finish   <<<512,  256, 0, stream>>>(part, b2, out, sumsq);
  k_outscale <<<512,  256, 0, stream>>>(out, sumsq);
}